// MLPPredictor_30202210026092
// MI455X (gfx1250) — compile-verified
//
#include <hip/hip_runtime.h>
#include <hip/hip_bf16.h>

typedef __attribute__((ext_vector_type(2))) float v2f;
typedef __attribute__((ext_vector_type(8))) float v8f;

#define D_DIM 128

// -------- Kernel 1: per-node partial scores via fp32 WMMA --------------------
// su[n] = dot(h[n], W[0:128]),  sv[n] = dot(h[n], W[128:256])
// One wave32 per 16-node tile; 32 x V_WMMA_F32_16X16X4_F32 accumulate K=128.
__global__ __launch_bounds__(256) void node_score_wmma(
    const float* __restrict__ h, const float* __restrict__ Ww,
    float* __restrict__ su, float* __restrict__ sv,
    int n_nodes, int n_tiles)
{
    const int lane = threadIdx.x & 31;
    const int wave = threadIdx.x >> 5;
    const int tile = blockIdx.x * (blockDim.x >> 5) + wave;
    if (tile >= n_tiles) return;            // wave-uniform: EXEC stays all-1s below

    const int half = lane >> 4;             // 0 -> K pair {0,1}; 1 -> K pair {2,3}
    const int sub  = lane & 15;             // A: M index within tile; B: N index

    // A-matrix row pointer (clamped for a possible ragged last tile; duplicate
    // rows are harmless since stores below are bounds-guarded).
    int m = tile * 16 + sub;
    if (m >= n_nodes) m = n_nodes - 1;
    const float* hrow = h + (size_t)m * D_DIM;

    // B-matrix: column 0 = Wu, column 1 = Wv, columns 2..15 = 0.
    const int   bvalid = (sub < 2) ? 1 : 0;
    const float* wrow  = Ww + (size_t)(bvalid ? sub : 0) * D_DIM;

    v8f acc = {};
    #pragma unroll
    for (int j = 0; j < D_DIM / 4; ++j) {
        const int k = j * 4 + half * 2;
        v2f a = *(const v2f*)(hrow + k);        // global_load_b64, 8B aligned
        v2f braw = *(const v2f*)(wrow + k);     // 1KB table: always cache-hot
        v2f b;
        b.x = bvalid ? braw.x : 0.0f;           // zero out unused B columns
        b.y = bvalid ? braw.y : 0.0f;
        // 8-arg form: (neg_a, A, neg_b, B, c_mod, C, reuse_a, reuse_b)
        acc = __builtin_amdgcn_wmma_f32_16x16x4_f32(
            false, a, false, b, (short)0, acc, false, false);
    }

    // D layout (ISA 7.12.2): VGPR r, lanes 0-15 -> (M=r, N=lane);
    // lanes 16-31 -> (M=r+8, N=lane-16).  Column N=0 => su, N=1 => sv.
    const int mbase = tile * 16 + half * 8;
    if (sub == 0) {
        #pragma unroll
        for (int r = 0; r < 8; ++r) {
            int mm = mbase + r;
            if (mm < n_nodes) su[mm] = acc[r];
        }
    } else if (sub == 1) {
        #pragma unroll
        for (int r = 0; r < 8; ++r) {
            int mm = mbase + r;
            if (mm < n_nodes) sv[mm] = acc[r];
        }
    }
}

// -------- Kernel 2: per-edge gather-add ------------------------------------
// out[e] = su[src[e]] + sv[dst[e]] + b.  su/sv (0.8 MB) are L2-resident.
__global__ __launch_bounds__(256) void edge_score(
    const float* __restrict__ su, const float* __restrict__ sv,
    const int* __restrict__ src, const int* __restrict__ dst,
    const float* __restrict__ Wb, float* __restrict__ out, int n_edges)
{
    const int e = blockIdx.x * blockDim.x + threadIdx.x;
    if (e < n_edges) {
        const int u = src[e];
        const int v = dst[e];
        out[e] = su[u] + sv[v] + Wb[0];
    }
}

// -------- Fallback (only if workspace is too small): direct per-edge dot ----
__global__ __launch_bounds__(256) void edge_score_direct(
    const float* __restrict__ h, const int* __restrict__ src,
    const int* __restrict__ dst, const float* __restrict__ Ww,
    const float* __restrict__ Wb, float* __restrict__ out, int n_edges)
{
    const int e = blockIdx.x * blockDim.x + threadIdx.x;
    if (e >= n_edges) return;
    const float* hu = h + (size_t)src[e] * D_DIM;
    const float* hv = h + (size_t)dst[e] * D_DIM;
    float s = Wb[0];
    #pragma unroll 8
    for (int k = 0; k < D_DIM; ++k)
        s += hu[k] * Ww[k] + hv[k] * Ww[D_DIM + k];
    out[e] = s;
}

extern "C" void kernel_launch(void* const* d_in, const int* in_sizes, int n_in,
                              void* d_out, int out_size, void* d_ws, size_t ws_size,
                              hipStream_t stream)
{
    const float* h   = (const float*)d_in[0];
    const int*   src = (const int*)d_in[1];
    const int*   dst = (const int*)d_in[2];
    const float* Ww  = (const float*)d_in[3];
    const float* Wb  = (const float*)d_in[4];
    float* out = (float*)d_out;

    const int n_nodes = in_sizes[0] / D_DIM;
    const int n_edges = in_sizes[1];

    const size_t need = (size_t)2 * (size_t)n_nodes * sizeof(float);
    if (ws_size >= need) {
        float* su = (float*)d_ws;
        float* sv = su + n_nodes;

        const int n_tiles = (n_nodes + 15) / 16;
        const int wpb = 8;                       // 8 waves (256 threads) per block
        dim3 grd1((n_tiles + wpb - 1) / wpb);
        node_score_wmma<<<grd1, dim3(256), 0, stream>>>(h, Ww, su, sv,
                                                        n_nodes, n_tiles);

        dim3 grd2((n_edges + 255) / 256);
        edge_score<<<grd2, dim3(256), 0, stream>>>(su, sv, src, dst, Wb,
                                                   out, n_edges);
    } else {
        dim3 grd((n_edges + 255) / 256);
        edge_score_direct<<<grd, dim3(256), 0, stream>>>(h, src, dst, Ww, Wb,
                                                         out, n_edges);
    }
}